// ResidueKpGNN_11106785427533
// MI455X (gfx1250) — compile-verified
//
#include <hip/hip_runtime.h>

#define NEG_GAT 0.2f
#define NEG_ACT 0.01f

typedef __attribute__((ext_vector_type(2))) float v2f;
typedef __attribute__((ext_vector_type(8))) float v8f;

__device__ __forceinline__ float leakyf(float x, float s) { return x > 0.0f ? x : s * x; }

// Sign-punned float atomic max. Works with memory initialized to 0xFFFFFFFF.
__device__ __forceinline__ void atomicMaxF(float* addr, float val) {
  if (__float_as_uint(val) >> 31) {
    atomicMin((unsigned int*)addr, __float_as_uint(val));
  } else {
    atomicMax((int*)addr, __float_as_int(val));
  }
}

// ---------------------------------------------------------------------------
// Node linear: XL = X@Wl + bl, XR = X@Wr + br using V_WMMA_F32_16X16X4_F32.
// FIN/FOUT are compile-time so the K-loop fully unrolls and tail guards are
// branchless clamped loads + cndmask (or vanish entirely when KPAD == FIN).
// A layout (16x4 f32): lanes 0-15 hold K=kk,kk+1 of row M=lane; lanes 16-31
// hold K=kk+2,kk+3 of row M=lane-16.  B (4x16): VGPR0 = {K=kk | K=kk+2},
// VGPR1 = {K=kk+1 | K=kk+3}, N = lane&15.  C/D: VGPR j -> M = j + 8*(lane>>4).
// ---------------------------------------------------------------------------
template <int FIN, int FOUT>
__global__ void node_linear_wmma_kernel(const float* __restrict__ X, int N,
                                        const float* __restrict__ Wl, const float* __restrict__ Bl,
                                        const float* __restrict__ Wr, const float* __restrict__ Br,
                                        float* __restrict__ XL, float* __restrict__ XR) {
  constexpr int KPAD = (FIN + 3) & ~3;
  constexpr int NSTEP = KPAD / 4;
  constexpr int NCT = (FOUT + 15) / 16;
  __shared__ float sWl[NCT * KPAD * 16];
  __shared__ float sWr[NCT * KPAD * 16];
  const int tid = threadIdx.x;
  const int lane = tid & 31;
  const int wave = tid >> 5;

  // Stage all weight column-tiles, zero padded, one barrier total.
  for (int i = tid; i < NCT * KPAD * 16; i += (int)blockDim.x) {
    const int n = i & 15;
    const int k = (i >> 4) % KPAD;
    const int ct = (i >> 4) / KPAD;
    const int col = ct * 16 + n;
    float wl = 0.0f, wr = 0.0f;
    if (k < FIN && col < FOUT) {
      wl = Wl[k * FOUT + col];
      wr = Wr[k * FOUT + col];
    }
    sWl[i] = wl;
    sWr[i] = wr;
  }
  __syncthreads();

  const int rowbase = (blockIdx.x * 4 + wave) * 16;
  if (rowbase >= N) return;  // wave-uniform: EXEC stays all-ones for WMMA

  int mrow = rowbase + (lane & 15);
  if (mrow > N - 1) mrow = N - 1;  // clamp: extra rows are skipped by the row < N store guard
  const float* xrow = X + (size_t)mrow * FIN;
  const int khalf = (lane >> 4) * 2;
  const int ncol = lane & 15;

  // Load A fragments once; reused for Wl, Wr and every column tile.
  v2f afrag[NSTEP];
#pragma unroll
  for (int t = 0; t < NSTEP; ++t) {
    const int k0 = t * 4 + khalf;
    if (KPAD == FIN) {
      afrag[t].x = xrow[k0];
      afrag[t].y = xrow[k0 + 1];
    } else {
      const int kax = (k0 < FIN) ? k0 : FIN - 1;
      const int kay = (k0 + 1 < FIN) ? k0 + 1 : FIN - 1;
      const float vx = xrow[kax];  // clamped: always in-bounds, no branch
      const float vy = xrow[kay];
      afrag[t].x = (k0 < FIN) ? vx : 0.0f;
      afrag[t].y = (k0 + 1 < FIN) ? vy : 0.0f;
    }
  }

#pragma unroll
  for (int ct = 0; ct < NCT; ++ct) {
    v8f accL = {};
    v8f accR = {};
#pragma unroll
    for (int t = 0; t < NSTEP; ++t) {
      const int k0 = t * 4 + khalf;
      const int bi = (ct * KPAD + k0) * 16 + ncol;
      v2f bl, br;
      bl.x = sWl[bi];
      bl.y = sWl[bi + 16];
      br.x = sWr[bi];
      br.y = sWr[bi + 16];
      accL = __builtin_amdgcn_wmma_f32_16x16x4_f32(false, afrag[t], false, bl, (short)0, accL,
                                                   false, false);
      accR = __builtin_amdgcn_wmma_f32_16x16x4_f32(false, afrag[t], false, br, (short)0, accR,
                                                   false, false);
    }
    const int col = ct * 16 + ncol;
    if (col < FOUT) {
      const float bLv = Bl[col];
      const float bRv = Br[col];
      const int mb = (lane >> 4) * 8;
#pragma unroll
      for (int j = 0; j < 8; ++j) {
        const int row = rowbase + mb + j;
        if (row < N) {
          XL[(size_t)row * FOUT + col] = accL[j] + bLv;
          XR[(size_t)row * FOUT + col] = accR[j] + bRv;
        }
      }
    }
  }
}

// ---------------------------------------------------------------------------
// Per-edge passes (H, C compile-time so feature loops fully unroll)
// ---------------------------------------------------------------------------
__global__ void edge_count_kernel(const int* __restrict__ dst, const float* __restrict__ ea, int E,
                                  float* __restrict__ cnt, float* __restrict__ easum) {
  int e = blockIdx.x * blockDim.x + threadIdx.x;
  if (e >= E) return;
  int d = dst[e];
  atomicAdd(&cnt[d], 1.0f);
  atomicAdd(&easum[d], ea[e]);
}

__global__ void loop_attr_kernel(const float* __restrict__ cnt, const float* __restrict__ easum,
                                 int N, float* __restrict__ la) {
  int n = blockIdx.x * blockDim.x + threadIdx.x;
  if (n >= N) return;
  la[n] = easum[n] / fmaxf(cnt[n], 1.0f);
}

template <int H, int C>
__global__ void edge_logit_kernel(const int* __restrict__ src, const int* __restrict__ dst,
                                  const float* __restrict__ ea, int E, const float* __restrict__ xl,
                                  const float* __restrict__ xr, const float* __restrict__ we,
                                  const float* __restrict__ att, float* __restrict__ elog,
                                  float* __restrict__ m) {
  constexpr int F = H * C;
  int e = blockIdx.x * blockDim.x + threadIdx.x;
  if (e >= E) return;
  int s = src[e], d = dst[e];
  float av = ea[e];
  const float* pl = xl + (size_t)s * F;
  const float* pr = xr + (size_t)d * F;
#pragma unroll
  for (int h = 0; h < H; ++h) {
    float lg = 0.0f;
#pragma unroll
    for (int c = 0; c < C; ++c) {
      int i = h * C + c;
      float v = leakyf(pl[i] + pr[i] + av * we[i], NEG_GAT);
      lg += v * att[i];
    }
    elog[(size_t)e * H + h] = lg;
    atomicMaxF(&m[(size_t)d * H + h], lg);
  }
}

template <int H, int C>
__global__ void self_logit_kernel(int N, const float* __restrict__ la, const float* __restrict__ xl,
                                  const float* __restrict__ xr, const float* __restrict__ we,
                                  const float* __restrict__ att, float* __restrict__ slog,
                                  float* __restrict__ m) {
  constexpr int F = H * C;
  int n = blockIdx.x * blockDim.x + threadIdx.x;
  if (n >= N) return;
  float av = la[n];
  const float* pl = xl + (size_t)n * F;
  const float* pr = xr + (size_t)n * F;
#pragma unroll
  for (int h = 0; h < H; ++h) {
    float lg = 0.0f;
#pragma unroll
    for (int c = 0; c < C; ++c) {
      int i = h * C + c;
      float v = leakyf(pl[i] + pr[i] + av * we[i], NEG_GAT);
      lg += v * att[i];
    }
    slog[(size_t)n * H + h] = lg;
    // runs after edge_logit_kernel on the same stream; one thread per node
    m[(size_t)n * H + h] = fmaxf(m[(size_t)n * H + h], lg);
  }
}

template <int H>
__global__ void edge_den_kernel(const int* __restrict__ dst, int E, const float* __restrict__ m,
                                float* __restrict__ elog, float* __restrict__ den) {
  int e = blockIdx.x * blockDim.x + threadIdx.x;
  if (e >= E) return;
  int d = dst[e];
#pragma unroll
  for (int h = 0; h < H; ++h) {
    float ex = expf(elog[(size_t)e * H + h] - m[(size_t)d * H + h]);
    elog[(size_t)e * H + h] = ex;
    atomicAdd(&den[(size_t)d * H + h], ex);
  }
}

template <int H>
__global__ void self_den_kernel(int N, const float* __restrict__ m, float* __restrict__ slog,
                                float* __restrict__ den) {
  int n = blockIdx.x * blockDim.x + threadIdx.x;
  if (n >= N) return;
#pragma unroll
  for (int h = 0; h < H; ++h) {
    float ex = expf(slog[(size_t)n * H + h] - m[(size_t)n * H + h]);
    slog[(size_t)n * H + h] = ex;
    den[(size_t)n * H + h] += ex;  // after edge_den_kernel; one thread per node
  }
}

template <int H, int C>
__global__ void edge_agg_kernel(const int* __restrict__ src, const int* __restrict__ dst, int E,
                                const float* __restrict__ elog, const float* __restrict__ den,
                                const float* __restrict__ xl, float* __restrict__ out) {
  constexpr int F = H * C;
  int e = blockIdx.x * blockDim.x + threadIdx.x;
  if (e >= E) return;
  int s = src[e], d = dst[e];
  const float* pl = xl + (size_t)s * F;
  float* po = out + (size_t)d * F;
#pragma unroll
  for (int h = 0; h < H; ++h) {
    float a = elog[(size_t)e * H + h] / den[(size_t)d * H + h];
#pragma unroll
    for (int c = 0; c < C; ++c) atomicAdd(&po[h * C + c], pl[h * C + c] * a);
  }
}

template <int H, int C>
__global__ void self_agg_kernel(int N, const float* __restrict__ slog,
                                const float* __restrict__ den, const float* __restrict__ xl,
                                float* __restrict__ out) {
  constexpr int F = H * C;
  int n = blockIdx.x * blockDim.x + threadIdx.x;
  if (n >= N) return;
  const float* pl = xl + (size_t)n * F;
  float* po = out + (size_t)n * F;
#pragma unroll
  for (int h = 0; h < H; ++h) {
    float a = slog[(size_t)n * H + h] / den[(size_t)n * H + h];
#pragma unroll
    for (int c = 0; c < C; ++c) po[h * C + c] += pl[h * C + c] * a;  // after edge_agg
  }
}

template <int F>
__global__ void finalize_kernel(float* __restrict__ h, const float* __restrict__ bias, int total) {
  int i = blockIdx.x * blockDim.x + threadIdx.x;
  if (i >= total) return;
  int col = i % F;
  h[i] = leakyf(h[i] + bias[col], NEG_ACT);
}

// ---------------------------------------------------------------------------
// Final edge MLP: [xc[src](34) | xc[dst](34) | ea(1)] -> 25 -> 10 -> 1, summed.
// ---------------------------------------------------------------------------
__global__ void edge_mlp_kernel(const int* __restrict__ src, const int* __restrict__ dst,
                                const float* __restrict__ ea, int E, const float* __restrict__ h3,
                                const float* __restrict__ h5, const float* __restrict__ w1,
                                const float* __restrict__ b1, const float* __restrict__ w2,
                                const float* __restrict__ b2, const float* __restrict__ w3,
                                const float* __restrict__ b3, float* __restrict__ out) {
  __shared__ float sw1[69 * 25];
  __shared__ float sw2[25 * 10];
  __shared__ float sb1[25];
  __shared__ float sb2[10];
  __shared__ float sw3[10];
  __shared__ float sb3;
  const int tid = threadIdx.x;
  for (int i = tid; i < 69 * 25; i += blockDim.x) sw1[i] = w1[i];
  for (int i = tid; i < 25 * 10; i += blockDim.x) sw2[i] = w2[i];
  if (tid < 25) sb1[tid] = b1[tid];
  if (tid < 10) {
    sb2[tid] = b2[tid];
    sw3[tid] = w3[tid];
  }
  if (tid == 0) sb3 = b3[0];
  __syncthreads();

  float ysum = 0.0f;
  for (int e = blockIdx.x * blockDim.x + tid; e < E; e += gridDim.x * blockDim.x) {
    int s = src[e], d = dst[e];
    float t1[25];
#pragma unroll
    for (int j = 0; j < 25; ++j) t1[j] = sb1[j];
    const float* ps3 = h3 + (size_t)s * 30;
    const float* ps5 = h5 + (size_t)s * 4;
    const float* pd3 = h3 + (size_t)d * 30;
    const float* pd5 = h5 + (size_t)d * 4;
    int k = 0;
    for (int i = 0; i < 30; ++i, ++k) {
      float v = ps3[i];
#pragma unroll
      for (int j = 0; j < 25; ++j) t1[j] += v * sw1[k * 25 + j];
    }
    for (int i = 0; i < 4; ++i, ++k) {
      float v = ps5[i];
#pragma unroll
      for (int j = 0; j < 25; ++j) t1[j] += v * sw1[k * 25 + j];
    }
    for (int i = 0; i < 30; ++i, ++k) {
      float v = pd3[i];
#pragma unroll
      for (int j = 0; j < 25; ++j) t1[j] += v * sw1[k * 25 + j];
    }
    for (int i = 0; i < 4; ++i, ++k) {
      float v = pd5[i];
#pragma unroll
      for (int j = 0; j < 25; ++j) t1[j] += v * sw1[k * 25 + j];
    }
    {
      float v = ea[e];
#pragma unroll
      for (int j = 0; j < 25; ++j) t1[j] += v * sw1[68 * 25 + j];
    }
    float t2[10];
#pragma unroll
    for (int jj = 0; jj < 10; ++jj) t2[jj] = sb2[jj];
#pragma unroll
    for (int j = 0; j < 25; ++j) {
      float v = leakyf(t1[j], NEG_ACT);
#pragma unroll
      for (int jj = 0; jj < 10; ++jj) t2[jj] += v * sw2[j * 10 + jj];
    }
    float y = sb3;
#pragma unroll
    for (int jj = 0; jj < 10; ++jj) y += leakyf(t2[jj], NEG_ACT) * sw3[jj];
    ysum += y;
  }
  // wave32 reduction + one atomic per wave
  for (int o = 16; o > 0; o >>= 1) ysum += __shfl_down(ysum, o, 32);
  if ((tid & 31) == 0) atomicAdd(out, ysum);
}

// ---------------------------------------------------------------------------
// Host orchestration
// ---------------------------------------------------------------------------
namespace {

struct Scratch {
  float *xl, *xr, *m, *den, *cnt, *easum, *la, *elog, *slog;
};

template <int FIN, int H, int C>
void gat_layer(const float* hin, int N, const int* ei, const float* ea, int E, const float* wl,
               const float* bl, const float* wr, const float* br, const float* we,
               const float* att, const float* bias, float* hout, const Scratch& ws,
               hipStream_t stream) {
  constexpr int F = H * C;
  const int* src = ei;
  const int* dst = ei + E;
  const int tb = 256;
  hipMemsetAsync(ws.cnt, 0, (size_t)N * sizeof(float), stream);
  hipMemsetAsync(ws.easum, 0, (size_t)N * sizeof(float), stream);
  hipMemsetAsync(ws.den, 0, (size_t)N * H * sizeof(float), stream);
  hipMemsetAsync(ws.m, 0xFF, (size_t)N * H * sizeof(float), stream);  // identity for punned max
  hipMemsetAsync(hout, 0, (size_t)N * F * sizeof(float), stream);

  edge_count_kernel<<<(E + tb - 1) / tb, tb, 0, stream>>>(dst, ea, E, ws.cnt, ws.easum);
  loop_attr_kernel<<<(N + tb - 1) / tb, tb, 0, stream>>>(ws.cnt, ws.easum, N, ws.la);

  node_linear_wmma_kernel<FIN, F>
      <<<(N + 63) / 64, 128, 0, stream>>>(hin, N, wl, bl, wr, br, ws.xl, ws.xr);

  edge_logit_kernel<H, C><<<(E + tb - 1) / tb, tb, 0, stream>>>(src, dst, ea, E, ws.xl, ws.xr, we,
                                                                att, ws.elog, ws.m);
  self_logit_kernel<H, C><<<(N + tb - 1) / tb, tb, 0, stream>>>(N, ws.la, ws.xl, ws.xr, we, att,
                                                                ws.slog, ws.m);
  edge_den_kernel<H><<<(E + tb - 1) / tb, tb, 0, stream>>>(dst, E, ws.m, ws.elog, ws.den);
  self_den_kernel<H><<<(N + tb - 1) / tb, tb, 0, stream>>>(N, ws.m, ws.slog, ws.den);
  edge_agg_kernel<H, C><<<(E + tb - 1) / tb, tb, 0, stream>>>(src, dst, E, ws.elog, ws.den, ws.xl,
                                                              hout);
  self_agg_kernel<H, C><<<(N + tb - 1) / tb, tb, 0, stream>>>(N, ws.slog, ws.den, ws.xl, hout);
  const int total = N * F;
  finalize_kernel<F><<<(total + tb - 1) / tb, tb, 0, stream>>>(hout, bias, total);
}

}  // namespace

extern "C" void kernel_launch(void* const* d_in, const int* in_sizes, int n_in, void* d_out,
                              int out_size, void* d_ws, size_t ws_size, hipStream_t stream) {
  const float* x = (const float*)d_in[0];
  const int* pep_ei = (const int*)d_in[1];
  const float* pep_ea = (const float*)d_in[2];
  const int* sp_ei = (const int*)d_in[3];
  const float* sp_ea = (const float*)d_in[4];
  const int* int_ei = (const int*)d_in[5];
  const float* int_ea = (const float*)d_in[6];

  const int N = in_sizes[0] / 64;
  const int EP = in_sizes[1] / 2;
  const int ES = in_sizes[3] / 2;
  const int EI = in_sizes[5] / 2;

  auto gp = [&](int layer, int j) { return (const float*)d_in[7 + layer * 7 + j]; };
  const float* e1w = (const float*)d_in[42];
  const float* e1b = (const float*)d_in[43];
  const float* e2w = (const float*)d_in[44];
  const float* e2b = (const float*)d_in[45];
  const float* e3w = (const float*)d_in[46];
  const float* e3b = (const float*)d_in[47];

  // Bump-allocate scratch from d_ws (256B aligned).
  char* base = (char*)d_ws;
  size_t off = 0;
  auto take = [&](size_t count) {
    float* p = (float*)(base + off);
    off += (count * sizeof(float) + 255) & ~(size_t)255;
    return p;
  };
  Scratch ws;
  ws.xl = take((size_t)N * 30);
  ws.xr = take((size_t)N * 30);
  ws.m = take((size_t)N * 3);
  ws.den = take((size_t)N * 3);
  ws.cnt = take((size_t)N);
  ws.easum = take((size_t)N);
  ws.la = take((size_t)N);
  size_t emax = (size_t)EP * 3;
  if ((size_t)ES * 2 > emax) emax = (size_t)ES * 2;
  ws.elog = take(emax);
  ws.slog = take((size_t)N * 3);
  float* h1 = take((size_t)N * 10);
  float* h2 = take((size_t)N * 15);
  float* h3 = take((size_t)N * 30);
  float* h4 = take((size_t)N * 4);
  float* h5 = take((size_t)N * 4);
  (void)ws_size;
  (void)n_in;
  (void)out_size;

  // layer dims: (c, heads) = (5,2),(5,3),(10,3),(2,2),(2,2); Fin = 64,10,15,30,4
  gat_layer<64, 2, 5>(x, N, pep_ei, pep_ea, EP, gp(0, 0), gp(0, 1), gp(0, 2), gp(0, 3), gp(0, 4),
                      gp(0, 5), gp(0, 6), h1, ws, stream);
  gat_layer<10, 3, 5>(h1, N, pep_ei, pep_ea, EP, gp(1, 0), gp(1, 1), gp(1, 2), gp(1, 3), gp(1, 4),
                      gp(1, 5), gp(1, 6), h2, ws, stream);
  gat_layer<15, 3, 10>(h2, N, pep_ei, pep_ea, EP, gp(2, 0), gp(2, 1), gp(2, 2), gp(2, 3), gp(2, 4),
                       gp(2, 5), gp(2, 6), h3, ws, stream);
  gat_layer<30, 2, 2>(h3, N, sp_ei, sp_ea, ES, gp(3, 0), gp(3, 1), gp(3, 2), gp(3, 3), gp(3, 4),
                      gp(3, 5), gp(3, 6), h4, ws, stream);
  gat_layer<4, 2, 2>(h4, N, sp_ei, sp_ea, ES, gp(4, 0), gp(4, 1), gp(4, 2), gp(4, 3), gp(4, 4),
                     gp(4, 5), gp(4, 6), h5, ws, stream);

  hipMemsetAsync(d_out, 0, sizeof(float), stream);
  edge_mlp_kernel<<<1024, 256, 0, stream>>>(int_ei, int_ei + EI, int_ea, EI, h3, h5, e1w, e1b, e2w,
                                            e2b, e3w, e3b, (float*)d_out);
}